// CoeffNet_58110907515322
// MI455X (gfx1250) — compile-verified
//
#include <hip/hip_runtime.h>
#include <math.h>

// ---------------------------------------------------------------------------
// CoeffNet (equivariant GNN) for MI455X / gfx1250, fp32 end-to-end.
// V_WMMA_F32_16X16X4_F32 for the dominant GEMM: coef = rad @ W[l1,l2,l3] + b.
// ---------------------------------------------------------------------------

typedef float v2f __attribute__((ext_vector_type(2)));
typedef float v8f __attribute__((ext_vector_type(8)));

#define FIN 16
#define FF  32      // num features
#define BB  32      // num basis
#define ROWF (9*FF) // 288 floats per edge feature row

// compile-time degree slicing of the 9-dim irrep axis
__host__ __device__ constexpr int S0c(int l) { return l == 0 ? 0 : (l == 1 ? 1 : 4); }
__host__ __device__ constexpr int SLc(int l) { return l == 0 ? 1 : (l == 1 ? 3 : 5); }

__device__ inline void sh9(float x, float y, float z, float* o) {
    const float c0  = 0.28209479177387814f;
    const float c1  = 0.4886025119029199f;
    const float c2  = 1.0925484305920792f;
    const float c20 = 0.31539156525252005f;
    const float c22 = 0.5462742152960396f;
    o[0] = c0;
    o[1] = c1 * y;  o[2] = c1 * z;  o[3] = c1 * x;
    o[4] = c2 * x * y;
    o[5] = c2 * y * z;
    o[6] = c20 * (3.0f * z * z - 1.0f);
    o[7] = c2 * x * z;
    o[8] = c22 * (x * x - y * y);
}

// ---------------------------------------------------------------------------
// 1) Gaunt coefficients G[9][9][9] via the reference's exact quadrature:
//    8-pt Gauss-Legendre in cos(theta) x 16-pt uniform in phi.
// ---------------------------------------------------------------------------
__global__ void gaunt_init(float* __restrict__ G) {
    __shared__ float Yg[128][9];
    __shared__ float wq[128];
    int t = threadIdx.x;
    if (t < 128) {
        const float gx[8] = {-0.9602898564975363f, -0.7966664774136267f,
                             -0.5255324099163290f, -0.1834346424956498f,
                              0.1834346424956498f,  0.5255324099163290f,
                              0.7966664774136267f,  0.9602898564975363f};
        const float gw[8] = { 0.10122853629037626f, 0.22238103445337447f,
                              0.31370664587788729f, 0.36268378337836199f,
                              0.36268378337836199f, 0.31370664587788729f,
                              0.22238103445337447f, 0.10122853629037626f};
        int i = t >> 4, j = t & 15;
        float ct  = gx[i];
        float st  = sqrtf(fmaxf(1.0f - ct * ct, 0.0f));
        float phi = (float)j * (6.283185307179586f / 16.0f);
        sh9(st * cosf(phi), st * sinf(phi), ct, Yg[t]);
        wq[t] = gw[i] * (6.283185307179586f / 16.0f);
    }
    __syncthreads();
    if (t < 729) {
        int a = t / 81, b = (t / 9) % 9, c = t % 9;
        float s = 0.0f;
        for (int q = 0; q < 128; ++q)
            s += Yg[q][a] * Yg[q][b] * Yg[q][c] * wq[q];
        G[t] = s;
    }
}

// ---------------------------------------------------------------------------
// 2) Per-edge geometry: SH(9) and radial basis (32)
// ---------------------------------------------------------------------------
__global__ void geom_kernel(const float* __restrict__ coords,
                            const int* __restrict__ dst, const int* __restrict__ src,
                            float* __restrict__ Ysh, float* __restrict__ rad, int E) {
    int e = blockIdx.x * blockDim.x + threadIdx.x;
    if (e >= E) return;
    int d = dst[e], s = src[e];
    float rx = coords[3*d+0] - coords[3*s+0];
    float ry = coords[3*d+1] - coords[3*s+1];
    float rz = coords[3*d+2] - coords[3*s+2];
    float r  = sqrtf(rx*rx + ry*ry + rz*rz + 1e-8f);
    float iv = 1.0f / r;
    float y9[9];
    sh9(rx*iv, ry*iv, rz*iv, y9);
    #pragma unroll
    for (int k = 0; k < 9; ++k) Ysh[(size_t)e*9 + k] = y9[k];
    #pragma unroll
    for (int b = 0; b < BB; ++b) {
        float c  = (4.0f / 31.0f) * (float)b;   // linspace(0,4,32)
        float dd = r - c;
        rad[(size_t)e*BB + b] = expf(-dd * dd);
    }
}

// ---------------------------------------------------------------------------
// 3) Embed: degree-wise dense 16 -> 32, bias only on the m==0 channel.
// ---------------------------------------------------------------------------
__global__ void embed_kernel(const float* __restrict__ xin,
                             const float* __restrict__ W0, const float* __restrict__ b0,
                             float* __restrict__ x, int E) {
    long long t = (long long)blockIdx.x * blockDim.x + threadIdx.x;
    int g = (int)(t & 31);
    long long em = t >> 5;
    if (em >= (long long)E * 9) return;
    int m = (int)(em % 9);
    int l = (m == 0) ? 0 : (m < 4 ? 1 : 2);
    const float* xr = xin + em * FIN;
    const float* w  = W0 + l * FIN * FF + g;
    float s = (m == 0) ? b0[g] : 0.0f;
    #pragma unroll
    for (int f = 0; f < FIN; ++f) s += xr[f] * w[f * FF];
    x[em * FF + g] = s;
}

__global__ void zero_kernel(float* __restrict__ p, long long n) {
    long long i = (long long)blockIdx.x * blockDim.x + threadIdx.x;
    if (i < n) p[i] = 0.0f;
}

__global__ void add_zero_kernel(float* __restrict__ x, float* __restrict__ msg, long long n) {
    long long i = (long long)blockIdx.x * blockDim.x + threadIdx.x;
    if (i < n) { x[i] += msg[i]; msg[i] = 0.0f; }
}

// ---------------------------------------------------------------------------
// Compile-time-specialized Gaunt contraction for one allowed (L1,L2,L3).
// acc[c] += coef * sum_{a in L1} xs9[a] * TT_L2[a][c]
// where TT_L2[a][c] = sum_{b in L2} Ysh[e,b] * G[a,b,c]  (precomputed, LDS).
// ---------------------------------------------------------------------------
template<int L1, int L2, int L3>
__device__ inline void contract_triple(int t, int e, int lane,
                                       const float* __restrict__ coefL,
                                       const float* __restrict__ ttL,
                                       const float xs9[9], float acc[9]) {
    constexpr int A0 = S0c(L1), DA = SLc(L1);
    constexpr int C0 = S0c(L3), DC = SLc(L3);
    const float cf = coefL[t * (16 * FF) + e * FF + lane];
    const float* tt = &ttL[(e * 3 + L2) * 81];
    #pragma unroll
    for (int ci = 0; ci < DC; ++ci) {
        float s = 0.0f;
        #pragma unroll
        for (int ai = 0; ai < DA; ++ai)
            s += xs9[A0 + ai] * tt[(A0 + ai) * 9 + (C0 + ci)];
        acc[C0 + ci] += cf * s;
    }
}

// ---------------------------------------------------------------------------
// 4) Message pass: one wave32 per 16-edge tile.
//    Phase A: TT[e][l2][a][c] = sum_b Ysh*G  (f-independent, LDS).
//    Phase B: 11 coef tiles (16x32) = rad_tile @ W[l1,l2,l3] + b via fully
//             unrolled V_WMMA_F32_16X16X4_F32 (8 K-steps x 2 N-subtiles),
//             bias preloaded into the C accumulator; parked in LDS.
//    Phase C: lane = feature f; loop edges; xs read coalesced from global
//             (L2-resident), TT reads wave-uniform broadcasts, accumulate
//             msg[e,:,f] in registers, one coalesced f32 atomic per element.
// ---------------------------------------------------------------------------
__global__ void __launch_bounds__(32)
mp_kernel(const float* __restrict__ x,
          const float* __restrict__ Ysh, const float* __restrict__ rad,
          const int* __restrict__ dst, const int* __restrict__ src,
          const float* __restrict__ Wmp, const float* __restrict__ bmp,
          const float* __restrict__ G, float* __restrict__ msg, int step) {
    __shared__ float Gl[729];
    __shared__ float yshL[16 * 9];
    __shared__ float radL[16 * BB];
    __shared__ float ttL[16 * 3 * 81];       // TT[e][l2][a*9+c]
    __shared__ float coefL[11 * 16 * FF];    // coef[t][e][f]
    __shared__ int   idxL[32];               // [0..15]=src, [16..31]=dst

    const int lane = threadIdx.x;
    const int base = blockIdx.x * 16;

    for (int t = lane; t < 729; t += 32) Gl[t] = G[t];
    if (lane < 16) {
        idxL[lane]      = src[base + lane];
        idxL[16 + lane] = dst[base + lane];
    }
    for (int t = lane; t < 16 * 9;  t += 32) yshL[t] = Ysh[(size_t)base * 9 + t];
    for (int t = lane; t < 16 * BB; t += 32) radL[t] = rad[(size_t)base * BB + t];
    __syncthreads();

    // warm L2/L0 for the gathered source rows used in phase C
    #pragma unroll
    for (int i = 0; i < 16; ++i)
        __builtin_prefetch(&x[(size_t)idxL[i] * ROWF + lane], 0, 0);

    // ---- Phase A: TT build (flat 16 edges x 3 l2 x 81 (a,c) = 3888 entries)
    for (int t = lane; t < 16 * 3 * 81; t += 32) {
        int e   = t / 243;
        int rem = t % 243;
        int l2  = rem / 81;
        int ac  = rem % 81;
        int a   = ac / 9, c = ac % 9;
        int q0  = (l2 == 0) ? 0 : (l2 == 1 ? 1 : 4);
        int db  = (l2 == 0) ? 1 : (l2 == 1 ? 3 : 5);
        float s = 0.0f;
        for (int b = 0; b < db; ++b)
            s += yshL[e * 9 + q0 + b] * Gl[a * 81 + (q0 + b) * 9 + c];
        ttL[t] = s;
    }

    // ---- Phase B: coef tiles via WMMA
    // A frag (16x4 f32): lane<16 -> M=lane,{K0,K1}; lane>=16 -> {K2,K3}.
    // B frag (4x16 f32): mirrored (lane&15 = N; lane>=16 -> K rows 2,3).
    // C (16x16 f32): reg r, lane l -> M = r + 8*(l>>4), N = l&15.
    {
        const int lo = lane & 15, hi = lane >> 4, koff = hi * 2;
        const float* Wbase = Wmp + (size_t)step * 27 * (BB * FF);
        const float* bbase = bmp + (size_t)step * 27 * FF;
        const int wl1[11] = {0,0,0, 1,1,1,1, 2,2,2,2};
        const int wl2[11] = {0,1,2, 0,1,1,2, 0,1,2,2};
        const int wl3[11] = {0,1,2, 1,0,2,1, 2,1,0,2};
        #pragma unroll
        for (int t3 = 0; t3 < 11; ++t3) {
            const int widx = wl1[t3] * 9 + wl2[t3] * 3 + wl3[t3];
            const float* Wt = Wbase + (size_t)widx * (BB * FF);
            const float* bt = bbase + (size_t)widx * FF;
            #pragma unroll
            for (int nb = 0; nb < 2; ++nb) {
                const int f = lo + nb * 16;
                const float bias = bt[f];
                v8f c = {bias, bias, bias, bias, bias, bias, bias, bias};
                #pragma unroll
                for (int kb = 0; kb < 8; ++kb) {
                    const int k = kb * 4 + koff;
                    v2f a, b;
                    a.x = radL[lo * BB + k];
                    a.y = radL[lo * BB + k + 1];
                    b.x = Wt[(size_t)k * FF + f];
                    b.y = Wt[(size_t)(k + 1) * FF + f];
                    c = __builtin_amdgcn_wmma_f32_16x16x4_f32(
                            false, a, false, b, (short)0, c, false, false);
                }
                #pragma unroll
                for (int r = 0; r < 8; ++r)
                    coefL[t3 * (16 * FF) + (r + 8 * hi) * FF + f] = c[r];
            }
        }
    }
    __syncthreads();

    // ---- Phase C: per-edge contraction, lane = output feature
    for (int e = 0; e < 16; ++e) {
        const int   sIdx = idxL[e];
        const int   dIdx = idxL[16 + e];
        const float* xsg = x + (size_t)sIdx * ROWF + lane;
        float xs9[9];
        #pragma unroll
        for (int a = 0; a < 9; ++a) xs9[a] = xsg[a * FF];
        float acc[9] = {0,0,0,0,0,0,0,0,0};

        contract_triple<0,0,0>( 0, e, lane, coefL, ttL, xs9, acc);
        contract_triple<0,1,1>( 1, e, lane, coefL, ttL, xs9, acc);
        contract_triple<0,2,2>( 2, e, lane, coefL, ttL, xs9, acc);
        contract_triple<1,0,1>( 3, e, lane, coefL, ttL, xs9, acc);
        contract_triple<1,1,0>( 4, e, lane, coefL, ttL, xs9, acc);
        contract_triple<1,1,2>( 5, e, lane, coefL, ttL, xs9, acc);
        contract_triple<1,2,1>( 6, e, lane, coefL, ttL, xs9, acc);
        contract_triple<2,0,2>( 7, e, lane, coefL, ttL, xs9, acc);
        contract_triple<2,1,1>( 8, e, lane, coefL, ttL, xs9, acc);
        contract_triple<2,2,0>( 9, e, lane, coefL, ttL, xs9, acc);
        contract_triple<2,2,2>(10, e, lane, coefL, ttL, xs9, acc);

        float* mrow = msg + (size_t)dIdx * ROWF + lane;
        #pragma unroll
        for (int c = 0; c < 9; ++c)
            atomicAdd(&mrow[c * FF], acc[c]);
    }
}

// ---------------------------------------------------------------------------
// 5) Final head: y1 = dense32(x) (+b1 on m=0), gate by scalar-channel sign,
//    out = dense1(y1*gate) (+b2 on m=0). One wave per edge, lane = feature,
//    wave32 shuffle reduction for the 32->1 projection.
// ---------------------------------------------------------------------------
__global__ void __launch_bounds__(256)
final_kernel(const float* __restrict__ x,
             const float* __restrict__ W1, const float* __restrict__ b1,
             const float* __restrict__ W2, const float* __restrict__ b2,
             float* __restrict__ out, int E) {
    __shared__ float w1L[3 * FF * FF];
    __shared__ float w2L[3 * FF];
    __shared__ float b1L[FF];
    int t = threadIdx.x;
    for (int i = t; i < 3 * FF * FF; i += 256) w1L[i] = W1[i];
    if (t < 3 * FF) w2L[t] = W2[t];
    if (t < FF)     b1L[t] = b1[t];
    __syncthreads();

    const float b2v  = b2[0];
    const int   wave = t >> 5, g = t & 31;
    const int   e    = blockIdx.x * 8 + wave;
    if (e >= E) return;
    const float* xr = x + (size_t)e * ROWF;

    float y[9];
    #pragma unroll
    for (int m = 0; m < 9; ++m) {
        const int l = (m == 0) ? 0 : (m < 4 ? 1 : 2);
        float s = (m == 0) ? b1L[g] : 0.0f;
        const float* w  = &w1L[l * FF * FF + g];
        const float* xm = xr + m * FF;
        #pragma unroll
        for (int f = 0; f < FF; ++f) s += xm[f] * w[f * FF];
        y[m] = s;
    }
    const float gate = (y[0] > 0.0f) ? 1.0f : 0.0f;
    #pragma unroll
    for (int m = 0; m < 9; ++m) {
        const int l = (m == 0) ? 0 : (m < 4 ? 1 : 2);
        float v = y[m] * gate * w2L[l * FF + g];
        #pragma unroll
        for (int off = 16; off >= 1; off >>= 1) v += __shfl_xor(v, off, 32);
        if (g == 0) out[(size_t)e * 9 + m] = v + ((m == 0) ? b2v : 0.0f);
    }
}

// ---------------------------------------------------------------------------
// Host launcher
// ---------------------------------------------------------------------------
extern "C" void kernel_launch(void* const* d_in, const int* in_sizes, int n_in,
                              void* d_out, int out_size, void* d_ws, size_t ws_size,
                              hipStream_t stream) {
    const float* x_dftb = (const float*)d_in[0];
    const float* coords = (const float*)d_in[1];
    const int*   dst    = (const int*)d_in[2];
    const int*   src    = (const int*)d_in[3];
    const float* W0     = (const float*)d_in[4];
    const float* b0     = (const float*)d_in[5];
    const float* Wmp    = (const float*)d_in[6];
    const float* bmp    = (const float*)d_in[7];
    const float* W1     = (const float*)d_in[8];
    const float* b1     = (const float*)d_in[9];
    const float* W2     = (const float*)d_in[10];
    const float* b2     = (const float*)d_in[11];
    float*       out    = (float*)d_out;

    const int E = in_sizes[2];             // edge count (== node count)

    char* ws = (char*)d_ws;
    size_t off = 0;
    auto carve = [&](size_t bytes) -> float* {
        float* p = (float*)(ws + off);
        off += (bytes + 255) & ~(size_t)255;
        return p;
    };
    float* G    = carve(729 * sizeof(float));
    float* Ysh  = carve((size_t)E * 9  * sizeof(float));
    float* rad  = carve((size_t)E * BB * sizeof(float));
    float* xbuf = carve((size_t)E * ROWF * sizeof(float));
    float* msg  = carve((size_t)E * ROWF * sizeof(float));
    (void)ws_size;

    const long long nFeat = (long long)E * ROWF;

    gaunt_init<<<1, 768, 0, stream>>>(G);
    geom_kernel<<<(E + 255) / 256, 256, 0, stream>>>(coords, dst, src, Ysh, rad, E);
    embed_kernel<<<(unsigned)((nFeat + 255) / 256), 256, 0, stream>>>(x_dftb, W0, b0, xbuf, E);
    zero_kernel<<<(unsigned)((nFeat + 255) / 256), 256, 0, stream>>>(msg, nFeat);

    for (int step = 0; step < 2; ++step) {
        mp_kernel<<<E / 16, 32, 0, stream>>>(xbuf, Ysh, rad, dst, src, Wmp, bmp, G, msg, step);
        add_zero_kernel<<<(unsigned)((nFeat + 255) / 256), 256, 0, stream>>>(xbuf, msg, nFeat);
    }

    final_kernel<<<(E + 7) / 8, 256, 0, stream>>>(xbuf, W1, b1, W2, b2, out, E);
}